// RoIPooling2D_79920751444220
// MI455X (gfx1250) — compile-verified
//
#include <hip/hip_runtime.h>
#include <cfloat>
#include <stdint.h>

#define C_TOT   256
#define FH      38
#define FW      38
#define OUTH    7
#define OUTW    7
#define CCHUNK  64
#define MAXD    13   // max staged region side: ceil(199/16)+1 = 13

typedef unsigned int u32x4 __attribute__((ext_vector_type(4)));
typedef int          i32x4 __attribute__((ext_vector_type(4)));
typedef int          i32x8 __attribute__((ext_vector_type(8)));

#if __has_builtin(__builtin_amdgcn_tensor_load_to_lds)
#define HAVE_TDM 1
#else
#define HAVE_TDM 0
#endif

__global__ __launch_bounds__(256)
void roipool_tdm_kernel(const float* __restrict__ feat,
                        const float* __restrict__ rois,
                        const int*   __restrict__ roi_idx,
                        float*       __restrict__ out)
{
    __shared__ float sreg[CCHUNK * MAXD * MAXD];   // 43.2 KB

    const int r  = blockIdx.x;
    const int c0 = blockIdx.y * CCHUNK;
    const int t  = threadIdx.x;

    // ROI box, quantized exactly like the reference: (int32)(roi * 1/16)
    const float4 rf = ((const float4*)rois)[r];
    const int x1 = (int)(rf.x * 0.0625f);
    const int y1 = (int)(rf.y * 0.0625f);
    const int x2 = (int)(rf.z * 0.0625f);
    const int y2 = (int)(rf.w * 0.0625f);
    const int w  = x2 - x1;
    const int h  = y2 - y1;
    const int bidx = roi_idx[r];

    // Staged region = ROI box clamped to the feature map (and LDS capacity)
    const int sx0 = max(x1, 0);
    const int sy0 = max(y1, 0);
    int sw = min(min(x2, FW) - sx0, MAXD); sw = max(sw, 0);
    int sh = min(min(y2, FH) - sy0, MAXD); sh = max(sh, 0);
    const int n = sw * sh;   // elements per channel

    const float* base = feat + ((size_t)bidx * C_TOT + c0) * (FH * FW);

#if HAVE_TDM
    // ---- Stage: one Tensor Data Mover DMA moves the whole (sw x sh x 64ch)
    //      tile into LDS, packed contiguously as [c][y][x]. ----
    if (t == 0 && n > 0) {
        const uint64_t ga       = (uint64_t)(uintptr_t)(base + sy0 * FW + sx0);
        const unsigned lds_base = (unsigned)(uintptr_t)sreg;

        // D# group 0: count=1 (valid), lds_addr, global_addr[56:0], type=2
        u32x4 g0;
        g0.x = 1u;                                            // count=1, user mode
        g0.y = lds_base;                                      // lds_addr (bytes)
        g0.z = (unsigned)(ga & 0xFFFFFFFFu);                  // global_addr[31:0]
        g0.w = (unsigned)((ga >> 32) & 0x01FFFFFFu)           // global_addr[56:32]
             | 0x80000000u;                                   // type=2 ("image")

        // D# group 1: data_size=4B; tensor dims == tile dims (no OOB); strides
        i32x8 g1;
        g1[0] = (2 << 16);                                    // data_size=2 (4B), mask=0
        g1[1] = (sw & 0xFFFF) << 16;                          // tensor_dim0[15:0]
        g1[2] = (sh & 0xFFFF) << 16;                          // dim0 hi=0 | tensor_dim1[15:0]
        g1[3] = (sw & 0xFFFF) << 16;                          // dim1 hi=0 | tile_dim0
        g1[4] = (sh & 0xFFFF) | (CCHUNK << 16);               // tile_dim1 | tile_dim2
        g1[5] = FW;                                           // tensor_dim0_stride[31:0] = 38
        g1[6] = ((FH * FW) & 0xFFFF) << 16;                   // stride0 hi=0 | stride1[15:0]=1444
        g1[7] = (FH * FW) >> 16;                              // tensor_dim1_stride[47:16] = 0

        // D# group 2: tensor_dim2 = 64 channels; higher dims unused
        i32x4 g2; g2.x = CCHUNK; g2.y = 0; g2.z = 0; g2.w = 0;
        // D# group 3: unused (dims 3/4)
        i32x4 g3; g3.x = 0; g3.y = 0; g3.z = 0; g3.w = 0;

#if defined(__clang_major__) && __clang_major__ >= 23
        i32x8 gpad; gpad[0]=0; gpad[1]=0; gpad[2]=0; gpad[3]=0;
        gpad[4]=0; gpad[5]=0; gpad[6]=0; gpad[7]=0;
        __builtin_amdgcn_tensor_load_to_lds(g0, g1, g2, g3, gpad, 0);
#else
        __builtin_amdgcn_tensor_load_to_lds(g0, g1, g2, g3, 0);
#endif
    }
#if __has_builtin(__builtin_amdgcn_s_wait_tensorcnt)
    __builtin_amdgcn_s_wait_tensorcnt(0);
#else
    asm volatile("s_wait_tensorcnt 0" ::: "memory");
#endif
    __syncthreads();

#else  // !HAVE_TDM — fall back to per-lane async global->LDS copies
    if (n > 0) {
        const int total = CCHUNK * n;
        for (int l = t; l < total; l += 256) {
            const int c   = l / n;
            const int rem = l - c * n;
            const int yy  = rem / sw;
            const int xx  = rem - yy * sw;
            const float* g = base + (size_t)c * (FH * FW) + (sy0 + yy) * FW + (sx0 + xx);
            const unsigned lds_off = (unsigned)(uintptr_t)(&sreg[l]);
            asm volatile("global_load_async_to_lds_b32 %0, %1, off"
                         :: "v"(lds_off), "v"(g) : "memory");
        }
    }
#if __has_builtin(__builtin_amdgcn_s_wait_asynccnt)
    __builtin_amdgcn_s_wait_asynccnt(0);
#else
    asm volatile("s_wait_asynccnt 0" ::: "memory");
#endif
    __syncthreads();
#endif

    // ---- Compute: 64 channels x 49 bins; flat index == flat output index -> coalesced stores ----
    const size_t obase = ((size_t)r * C_TOT + c0) * (OUTH * OUTW);
    for (int o = t; o < CCHUNK * OUTH * OUTW; o += 256) {
        const int c = o / (OUTH * OUTW);
        const int b = o - c * (OUTH * OUTW);
        const int i = b / OUTW;
        const int j = b - i * OUTW;

        // Reference bin bounds: b0 = start + floor(k*len/7), b1 = start + ceil((k+1)*len/7)
        int by0 = y1 + (i * h) / OUTH;
        int by1 = y1 + ((i + 1) * h + OUTH - 1) / OUTH;
        int bx0 = x1 + (j * w) / OUTW;
        int bx1 = x1 + ((j + 1) * w + OUTW - 1) / OUTW;

        // Intersect with the staged (in-bounds) region
        by0 = max(by0, sy0); by1 = min(by1, sy0 + sh);
        bx0 = max(bx0, sx0); bx1 = min(bx1, sx0 + sw);

        float m = -FLT_MAX;   // == jnp.finfo(float32).min fill value
        const float* sc = &sreg[c * n];
        for (int yy = by0; yy < by1; ++yy) {
            const float* row = sc + (yy - sy0) * sw - sx0;
            for (int xx = bx0; xx < bx1; ++xx)
                m = fmaxf(m, row[xx]);
        }
        out[obase + o] = m;
    }
}

extern "C" void kernel_launch(void* const* d_in, const int* in_sizes, int n_in,
                              void* d_out, int out_size, void* d_ws, size_t ws_size,
                              hipStream_t stream) {
    const float* feat = (const float*)d_in[0];
    const float* rois = (const float*)d_in[1];
    const int*   ridx = (const int*)d_in[2];
    float*       out  = (float*)d_out;

    const int R = in_sizes[2];                 // 128 ROIs
    dim3 grid(R, C_TOT / CCHUNK);              // (128, 4)
    roipool_tdm_kernel<<<grid, 256, 0, stream>>>(feat, rois, ridx, out);
}